// Attn_88691074662550
// MI455X (gfx1250) — compile-verified
//
#include <hip/hip_runtime.h>
#include <hip/hip_bf16.h>
#include <cmath>

// Problem constants
#define SEQ  262144
#define HID  256
#define TILE_ROWS 128            // rows per workgroup in the GEMV kernel
#define PITCH (HID + 4)          // LDS row pitch (dwords): +4 keeps 16B align, kills bank conflicts
#define NBLK (SEQ / TILE_ROWS)   // 2048 score blocks

typedef __attribute__((ext_vector_type(2))) float v2f;
typedef __attribute__((ext_vector_type(4))) float v4f;
typedef __attribute__((ext_vector_type(8))) float v8f;

// ---------------------------------------------------------------------------
// K1: v[h] = sum_k hidden[k]*W[k,h];  c = hidden·b     (one block, 256 thr)
// ---------------------------------------------------------------------------
__global__ __launch_bounds__(256) void k_prep(const float* __restrict__ hidden,
                                              const float* __restrict__ W,
                                              const float* __restrict__ b,
                                              float* __restrict__ ws) {
  const int t = threadIdx.x;
  float acc = 0.f;
#pragma unroll 8
  for (int k = 0; k < HID; ++k)
    acc = fmaf(hidden[k], W[k * HID + t], acc);   // coalesced: column t of W
  ws[t] = acc;

  __shared__ float red[HID];
  red[t] = hidden[t] * b[t];
  __syncthreads();
  for (int off = HID / 2; off > 0; off >>= 1) {
    if (t < off) red[t] += red[t + off];
    __syncthreads();
  }
  if (t == 0) ws[HID] = red[0];
}

// ---------------------------------------------------------------------------
// K2: scores[s] = enc_s · v + c  via WMMA f32 16x16x4, plus per-block
//     softmax stats (max, sum-exp). 2048 blocks x 256 threads (8 waves).
// ---------------------------------------------------------------------------
__global__ __launch_bounds__(256) void k_scores(const float* __restrict__ enc,
                                                const float* __restrict__ ws,
                                                float* __restrict__ out,
                                                float* __restrict__ bmax,
                                                float* __restrict__ bsum) {
  __shared__ float tile[TILE_ROWS * PITCH];   // ~130 KB staged enc tile
  __shared__ float vsh[HID];
  __shared__ float ssh[TILE_ROWS];
  __shared__ float red[256];

  const int    t    = threadIdx.x;
  const int    blk  = blockIdx.x;
  const size_t row0 = (size_t)blk * TILE_ROWS;

  vsh[t] = ws[t];                              // stage v

  // Stage 128x256 f32 tile with fully coalesced b128 loads.
  const v4f* src = (const v4f*)(enc + row0 * HID);
#pragma unroll
  for (int i = t; i < TILE_ROWS * (HID / 4); i += 256) {
    const int r  = i >> 6;                     // /(HID/4)
    const int c4 = i & 63;
    v4f g = src[i];
    *(v4f*)&tile[r * PITCH + c4 * 4] = g;
  }
  __syncthreads();

  // Each wave computes 16 rows. A = enc tile (16x4 per step, LDS, conflict-free:
  // lane L%16 picks row -> bank stride 4; lane/16 picks K-pair {0,1}/{2,3}).
  // B = v chunk broadcast across all 16 columns -> D[m][*] = dot(enc_row_m, v).
  const int   wave  = t >> 5;
  const int   lane  = t & 31;
  const int   mrow  = lane & 15;
  const int   khalf = (lane >> 4) * 2;         // 0 or 2
  const float* arow = &tile[(wave * 16 + mrow) * PITCH + khalf];
  const float* brow = &vsh[khalf];

  v8f acc = {};
#pragma unroll 8
  for (int k0 = 0; k0 < HID; k0 += 4) {        // 64 x v_wmma_f32_16x16x4_f32
    v2f a = *(const v2f*)(arow + k0);
    v2f b = *(const v2f*)(brow + k0);
    acc = __builtin_amdgcn_wmma_f32_16x16x4_f32(
        false, a, false, b, (short)0, acc, false, false);
  }

  // D layout: VGPR r, lanes 0-15 -> (M=r, N=lane); lanes 16-31 -> (M=8+r).
  // All N columns identical; lanes 0 and 16 publish the 16 row scores.
  const float c = ws[HID];
  if (lane == 0) {
#pragma unroll
    for (int r = 0; r < 8; ++r) ssh[wave * 16 + r] = acc[r] + c;
  } else if (lane == 16) {
#pragma unroll
    for (int r = 0; r < 8; ++r) ssh[wave * 16 + 8 + r] = acc[r] + c;
  }
  __syncthreads();

  // Park raw scores in d_out; compute block max and sum-exp.
  float s = (t < TILE_ROWS) ? ssh[t] : -INFINITY;
  if (t < TILE_ROWS) out[row0 + t] = s;
  red[t] = s;
  __syncthreads();
  for (int off = 128; off > 0; off >>= 1) {
    if (t < off) red[t] = fmaxf(red[t], red[t + off]);
    __syncthreads();
  }
  const float m = red[0];
  __syncthreads();
  red[t] = (t < TILE_ROWS) ? expf(s - m) : 0.f;
  __syncthreads();
  for (int off = 128; off > 0; off >>= 1) {
    if (t < off) red[t] += red[t + off];
    __syncthreads();
  }
  if (t == 0) { bmax[blk] = m; bsum[blk] = red[0]; }
}

// ---------------------------------------------------------------------------
// K3: merge 2048 block stats: M = max m_b ; S = sum s_b * exp(m_b - M)
// ---------------------------------------------------------------------------
__global__ __launch_bounds__(256) void k_reduce(float* __restrict__ ws,
                                                const float* __restrict__ bmax,
                                                const float* __restrict__ bsum) {
  __shared__ float red[256];
  const int t = threadIdx.x;
  float m = -INFINITY;
  for (int i = t; i < NBLK; i += 256) m = fmaxf(m, bmax[i]);
  red[t] = m;
  __syncthreads();
  for (int off = 128; off > 0; off >>= 1) {
    if (t < off) red[t] = fmaxf(red[t], red[t + off]);
    __syncthreads();
  }
  const float M = red[0];
  __syncthreads();
  float s = 0.f;
  for (int i = t; i < NBLK; i += 256) s += bsum[i] * expf(bmax[i] - M);
  red[t] = s;
  __syncthreads();
  for (int off = 128; off > 0; off >>= 1) {
    if (t < off) red[t] += red[t + off];
    __syncthreads();
  }
  if (t == 0) { ws[HID + 1] = M; ws[HID + 2] = red[0]; }
}

// ---------------------------------------------------------------------------
// K4: out[i] = exp(score[i] - M) / S      (1 MB in-place pass over d_out)
// ---------------------------------------------------------------------------
__global__ __launch_bounds__(256) void k_norm(float* __restrict__ out,
                                              const float* __restrict__ ws) {
  const int i = blockIdx.x * 256 + threadIdx.x;
  const float M    = ws[HID + 1];
  const float invS = 1.0f / ws[HID + 2];
  out[i] = expf(out[i] - M) * invS;
}

// ---------------------------------------------------------------------------
extern "C" void kernel_launch(void* const* d_in, const int* in_sizes, int n_in,
                              void* d_out, int out_size, void* d_ws, size_t ws_size,
                              hipStream_t stream) {
  const float* hidden = (const float*)d_in[0];
  const float* enc    = (const float*)d_in[1];
  const float* W      = (const float*)d_in[2];
  const float* b      = (const float*)d_in[3];
  float*       out    = (float*)d_out;

  // ws layout (floats): [0..255]=v, [256]=c, [257]=M, [258]=S,
  //                     [260..260+NBLK)=block max, then NBLK block sums.
  float* ws   = (float*)d_ws;
  float* bmax = ws + 260;
  float* bsum = ws + 260 + NBLK;

  k_prep  <<<1,        256, 0, stream>>>(hidden, W, b, ws);
  k_scores<<<NBLK,     256, 0, stream>>>(enc, ws, out, bmax, bsum);
  k_reduce<<<1,        256, 0, stream>>>(ws, bmax, bsum);
  k_norm  <<<SEQ/256,  256, 0, stream>>>(out, ws);
}